// iResNet_89713276879568
// MI455X (gfx1250) — compile-verified
//
#include <hip/hip_runtime.h>

typedef __attribute__((ext_vector_type(2))) float v2f;
typedef __attribute__((ext_vector_type(4))) float f4;
typedef __attribute__((ext_vector_type(8))) float v8f;

#define CC   64            // channels
#define HH   256
#define WW   512
#define NL   81            // 2*40+1 disparity levels
#define CS   (HH * WW)     // channel stride in floats
#define BSTR 616           // LDS row stride: 48 + 512 + 48 = 608 data, pad to 616
                           // (2*616 mod 64 == 16 -> lane-halves hit disjoint banks)

// corr1d cost volume via fp32 WMMA 16x16x4.
// One block per (b,h). conv2b row-strip staged in LDS with zeroed +/-48 halo.
__global__ __launch_bounds__(256)
void corr1d_wmma(const float* __restrict__ A, const float* __restrict__ Bm,
                 float* __restrict__ out) {
  extern __shared__ float lds[];  // [CC][BSTR]

  const int tid = threadIdx.x;
  const int bh  = blockIdx.x;
  const int b   = bh >> 8;            // H = 256
  const int h   = bh & (HH - 1);

  const float* base_a = A  + (size_t)b * CC * CS + (size_t)h * WW;
  const float* base_b = Bm + (size_t)b * CC * CS + (size_t)h * WW;

  // ---- zero the halo columns of B_lds (48 left + 48 right per row = 24 float4)
  for (int i = tid; i < CC * 24; i += 256) {
    int c = i / 24, j = i % 24;
    int col4 = (j < 12) ? j : (j + 128);        // right halo starts at float4 140
    f4 z = {0.0f, 0.0f, 0.0f, 0.0f};
    *(f4*)(&lds[c * BSTR + col4 * 4]) = z;
  }
  // ---- stage conv2b[b, :, h, :] into LDS (64 rows x 128 float4)
  for (int i = tid; i < CC * (WW / 4); i += 256) {
    int c = i >> 7, q = i & 127;
    f4 v = *(const f4*)(base_b + (size_t)c * CS + q * 4);
    *(f4*)(&lds[c * BSTR + 48 + q * 4]) = v;
  }
  __syncthreads();

  const int wave = tid >> 5;
  const int lane = tid & 31;
  const int hl   = lane >> 4;   // lane half (0: lanes 0-15, 1: lanes 16-31)
  const int pos  = lane & 15;

  const float scale = 1.0f / 64.0f;

  for (int it = 0; it < 4; ++it) {
    const int wb = (wave + it * 8) << 4;   // this wave's w-block base (0..496)

    v8f acc[7] = {};

    #pragma unroll 2
    for (int k0 = 0; k0 < CC; k0 += 4) {
      // A fragment (16x4 f32): reg v holds A[M=pos][k = k0 + 2*hl + v]
      const float* ap = base_a + (size_t)(k0 + 2 * hl) * CS + (wb + pos);
      v2f af;
      af.x = ap[0];
      af.y = ap[CS];

      // B fragments from LDS: reg v holds B[k = k0 + 2*hl + v][N = pos]
      const float* bp0 = &lds[(k0 + 2 * hl) * BSTR + wb + pos]; // = halo(48) + (wb-48) + pos
      #pragma unroll
      for (int t = 0; t < 7; ++t) {
        const float* bp = bp0 + t * 16;
        v2f bf;
        bf.x = bp[0];
        bf.y = bp[BSTR];
        acc[t] = __builtin_amdgcn_wmma_f32_16x16x4_f32(
            false, af, false, bf, (short)0, acc[t], false, false);
      }
    }

    // D layout: reg r holds (M = r + 8*hl, N = pos); disparity d = 16t + N - M - 8
    #pragma unroll
    for (int t = 0; t < 7; ++t) {
      #pragma unroll
      for (int r = 0; r < 8; ++r) {
        int M = r + 8 * hl;
        int d = 16 * t + pos - M - 8;
        if (d >= 0 && d < NL) {
          size_t o = (((size_t)b * NL + d) * HH + h) * WW + (size_t)(wb + M);
          out[o] = acc[t][r] * scale;
        }
      }
    }
  }
}

extern "C" void kernel_launch(void* const* d_in, const int* in_sizes, int n_in,
                              void* d_out, int out_size, void* d_ws, size_t ws_size,
                              hipStream_t stream) {
  const float* a  = (const float*)d_in[0];   // conv2a [4,64,256,512] f32
  const float* bm = (const float*)d_in[1];   // conv2b [4,64,256,512] f32
  float* out = (float*)d_out;                // [4,81,256,512] f32

  dim3 grid(4 * HH);   // one block per (b,h)
  dim3 block(256);
  size_t shmem = (size_t)CC * BSTR * sizeof(float);  // 157,696 B
  corr1d_wmma<<<grid, block, shmem, stream>>>(a, bm, out);
}